// GroundingLoss_15977278341546
// MI455X (gfx1250) — compile-verified
//
#include <hip/hip_runtime.h>
#include <math.h>

// CDNA5 / gfx1250: wave32, native-f32 WMMA 16x16x4 path.
typedef __attribute__((ext_vector_type(2))) float v2f;
typedef __attribute__((ext_vector_type(4))) float v4f;
typedef __attribute__((ext_vector_type(8))) float v8f;

#define ROWPAD 260   // 256 + 4: lane bank stride = 4 -> conflict-free ds_load_b64

// Column-wise (over i) softmax * sim reduction for one 32x32 sim tile held as
// 4 accumulators. Returns the wave-summed value (each column counted twice).
__device__ __forceinline__ float softmax_reduce(const v8f& c00, const v8f& c01,
                                                const v8f& c10, const v8f& c11) {
  float mlo = -INFINITY, mhi = -INFINITY;
#pragma unroll
  for (int v = 0; v < 8; ++v) {
    mlo = fmaxf(mlo, fmaxf(c00[v], c10[v]));   // columns 0..15 family
    mhi = fmaxf(mhi, fmaxf(c01[v], c11[v]));   // columns 16..31 family
  }
  mlo = fmaxf(mlo, __shfl_xor(mlo, 16, 32));
  mhi = fmaxf(mhi, __shfl_xor(mhi, 16, 32));

  float nlo = 0.f, dlo = 0.f, nhi = 0.f, dhi = 0.f;
#pragma unroll
  for (int v = 0; v < 8; ++v) {
    float e;
    e = __expf(c00[v] - mlo); dlo += e; nlo += e * c00[v];
    e = __expf(c10[v] - mlo); dlo += e; nlo += e * c10[v];
    e = __expf(c01[v] - mhi); dhi += e; nhi += e * c01[v];
    e = __expf(c11[v] - mhi); dhi += e; nhi += e * c11[v];
  }
  nlo += __shfl_xor(nlo, 16, 32); dlo += __shfl_xor(dlo, 16, 32);
  nhi += __shfl_xor(nhi, 16, 32); dhi += __shfl_xor(dhi, 16, 32);

  float val = nlo / dlo + nhi / dhi;
#pragma unroll
  for (int off = 16; off >= 1; off >>= 1)
    val += __shfl_xor(val, off, 32);
  return val;   // = 2 * sum_j colval_j
}

// ---------------------------------------------------------------------------
// Kernel 1: block = one 'a' x 16 b's. x[a] staged in LDS (shared by 8 waves),
// each wave register-blocks 2 b's: per k-step 2 ds_load_b64 + 4 global_load_b64
// feed 8 v_wmma_f32_16x16x4_f32.
// ---------------------------------------------------------------------------
__global__ __launch_bounds__(256) void grounding_sim_kernel(
    const float* __restrict__ x,   // [256][32][256]
    const float* __restrict__ z,   // [256][32][256]
    float* __restrict__ S)         // [256][256]
{
  __shared__ float xs[32 * ROWPAD];

  const int tid  = threadIdx.x;
  const int wave = tid >> 5;
  const int lane = tid & 31;

  const int a  = blockIdx.x >> 4;
  const int bg = (blockIdx.x & 15) << 4;
  const int b0 = bg + (wave << 1);
  const int b1 = b0 + 1;

  const float* xa = x + (size_t)a * (32 * 256);

  // Cooperative stage of x[a] -> LDS (row-major, padded rows).
#pragma unroll
  for (int i = 0; i < 8; ++i) {
    int idx = i * 1024 + tid * 4;          // 4 floats per thread per iter
    int row = idx >> 8;
    int col = idx & 255;
    v4f v = *(const v4f*)(xa + idx);
    *(v4f*)(xs + row * ROWPAD + col) = v;
  }
  __syncthreads();

  // Fragment addressing (ISA 7.12.2, 32-bit A 16x4 / B 4x16):
  //   lanes 0-15 : row/col = lane,    K = {k0, k0+1}
  //   lanes 16-31: row/col = lane-16, K = {k0+2, k0+3}
  const int r  = lane & 15;
  const int ko = (lane >> 4) << 1;

  const float* pa0 = xs + r * ROWPAD + ko;           // LDS: rows 0..15
  const float* pa1 = pa0 + 16 * ROWPAD;              // LDS: rows 16..31
  const float* pb0_0 = z + (size_t)b0 * (32 * 256) + r * 256 + ko;
  const float* pb1_0 = pb0_0 + 16 * 256;
  const float* pb0_1 = z + (size_t)b1 * (32 * 256) + r * 256 + ko;
  const float* pb1_1 = pb0_1 + 16 * 256;

  v8f cA00 = {}, cA01 = {}, cA10 = {}, cA11 = {};    // pair (a, b0)
  v8f cB00 = {}, cB01 = {}, cB10 = {}, cB11 = {};    // pair (a, b1)

#pragma unroll 4
  for (int k0 = 0; k0 < 256; k0 += 4) {
    v2f fa0 = *(const v2f*)(pa0 + k0);               // ds_load_b64
    v2f fa1 = *(const v2f*)(pa1 + k0);
    v2f g00 = *(const v2f*)(pb0_0 + k0);             // global_load_b64
    v2f g10 = *(const v2f*)(pb1_0 + k0);
    v2f g01 = *(const v2f*)(pb0_1 + k0);
    v2f g11 = *(const v2f*)(pb1_1 + k0);

    cA00 = __builtin_amdgcn_wmma_f32_16x16x4_f32(false, fa0, false, g00,
                                                 (short)0, cA00, false, false);
    cA01 = __builtin_amdgcn_wmma_f32_16x16x4_f32(false, fa0, false, g10,
                                                 (short)0, cA01, false, false);
    cA10 = __builtin_amdgcn_wmma_f32_16x16x4_f32(false, fa1, false, g00,
                                                 (short)0, cA10, false, false);
    cA11 = __builtin_amdgcn_wmma_f32_16x16x4_f32(false, fa1, false, g10,
                                                 (short)0, cA11, false, false);
    cB00 = __builtin_amdgcn_wmma_f32_16x16x4_f32(false, fa0, false, g01,
                                                 (short)0, cB00, false, false);
    cB01 = __builtin_amdgcn_wmma_f32_16x16x4_f32(false, fa0, false, g11,
                                                 (short)0, cB01, false, false);
    cB10 = __builtin_amdgcn_wmma_f32_16x16x4_f32(false, fa1, false, g01,
                                                 (short)0, cB10, false, false);
    cB11 = __builtin_amdgcn_wmma_f32_16x16x4_f32(false, fa1, false, g11,
                                                 (short)0, cB11, false, false);
  }

  // Fused softmax-weighted reduction; each column duplicated in 2 lanes:
  // S = total / (2 * 32).
  float vA = softmax_reduce(cA00, cA01, cA10, cA11);
  float vB = softmax_reduce(cB00, cB01, cB10, cB11);

  if (lane == 0) {
    S[a * 256 + b0] = vA * (1.0f / 64.0f);
    S[a * 256 + b1] = vB * (1.0f / 64.0f);
  }
}

// ---------------------------------------------------------------------------
// Kernel 2: symmetric InfoNCE on S [256x256].
//   loss = mean_n ( LSE(S[:,n]) + LSE(S[n,:]) - 2*S[n,n] )
// ---------------------------------------------------------------------------
__global__ __launch_bounds__(256) void grounding_loss_kernel(
    const float* __restrict__ S, float* __restrict__ out)
{
  const int n = threadIdx.x;
  float mr = -INFINITY, mc = -INFINITY;
  for (int j = 0; j < 256; ++j) {
    mr = fmaxf(mr, S[n * 256 + j]);
    mc = fmaxf(mc, S[j * 256 + n]);
  }
  float sr = 0.f, sc = 0.f;
  for (int j = 0; j < 256; ++j) {
    sr += __expf(S[n * 256 + j] - mr);
    sc += __expf(S[j * 256 + n] - mc);
  }
  const float diag = S[n * 257];
  float val = (mr + logf(sr)) + (mc + logf(sc)) - 2.0f * diag;

  __shared__ float red[256];
  red[n] = val;
  __syncthreads();
  for (int s = 128; s > 0; s >>= 1) {
    if (n < s) red[n] += red[n + s];
    __syncthreads();
  }
  if (n == 0) out[0] = red[0] * (1.0f / 256.0f);
}

// ---------------------------------------------------------------------------
extern "C" void kernel_launch(void* const* d_in, const int* in_sizes, int n_in,
                              void* d_out, int out_size, void* d_ws, size_t ws_size,
                              hipStream_t stream) {
  const float* x = (const float*)d_in[0];   // [256,32,256] f32
  const float* z = (const float*)d_in[1];   // [256,32,256] f32
  float* S   = (float*)d_ws;                // 256*256*4 = 256 KB scratch
  float* out = (float*)d_out;               // scalar f32

  // 4096 blocks: each = one a x 16 b's; 8 waves x 2 b's per wave.
  grounding_sim_kernel<<<4096, 256, 0, stream>>>(x, z, S);
  grounding_loss_kernel<<<1, 256, 0, stream>>>(S, out);
}